// GP_64622077935834
// MI455X (gfx1250) — compile-verified
//
#include <hip/hip_runtime.h>
#include <math.h>

// Problem dims (fixed by the reference)
#define GN 4096
#define GD 128
#define GM 512
#define GL 32

#define KC 64      // k-chunk staged in LDS
#define LDSTR 66   // padded LDS row stride (dwords): 66*4=264B == 2 banks/row step

typedef __attribute__((ext_vector_type(2))) float v2f;
typedef __attribute__((ext_vector_type(8))) float v8f;

// V_WMMA_F32_16X16X4_F32: D(16x16 f32) = A(16x4 f32) * B(4x16 f32) + C
// A/B: 2 VGPRs per lane; row/col = lane%16, K = 2*(lane/16)+v
// C/D: 8 VGPRs per lane; M = i + 8*(lane/16), N = lane%16
__device__ __forceinline__ v8f wmma4(v2f a, v2f b, v8f c) {
  return __builtin_amdgcn_wmma_f32_16x16x4_f32(
      /*neg_a=*/false, a, /*neg_b=*/false, b,
      /*c_mod=*/(short)0, c, /*reuse_a=*/false, /*reuse_b=*/false);
}

// Async global->LDS copy of 16 bytes (VGLOBAL encoding, ASYNCcnt tracked).
__device__ __forceinline__ void async_copy_b128(uint32_t lds_off, const void* gptr) {
  uint64_t g = (uint64_t)(uintptr_t)gptr;
  asm volatile("global_load_async_to_lds_b128 %0, %1, off"
               :: "v"(lds_off), "v"(g) : "memory");
}
__device__ __forceinline__ void async_wait0() {
  asm volatile("s_wait_asynccnt 0x0" ::: "memory");
}

// ---------------------------------------------------------------- row norms
__global__ void row_norms(const float* __restrict__ A, float* __restrict__ out, int rows) {
  int r = blockIdx.x * blockDim.x + threadIdx.x;
  if (r >= rows) return;
  const float* a = A + (size_t)r * GD;
  float s = 0.f;
#pragma unroll 8
  for (int k = 0; k < GD; ++k) s += a[k] * a[k];
  out[r] = s;
}

// --------------------------------------------- RBF kernel matrix via f32 WMMA
__global__ void rbf_wmma(const float* __restrict__ A, const float* __restrict__ An,
                         const float* __restrict__ B, const float* __restrict__ Bn,
                         const float* __restrict__ ls, float* __restrict__ K,
                         int ldk) {
  const int wave = threadIdx.x >> 5, lane = threadIdx.x & 31;
  const int r = lane & 15, h = lane >> 4;
  const int m0 = blockIdx.y * 16;
  const int n0 = blockIdx.x * 128 + wave * 16;
  const float* arow = A + (size_t)(m0 + r) * GD;
  const float* brow = B + (size_t)(n0 + r) * GD;
  v8f acc = {};
  for (int k = 0; k < GD; k += 4) {
    v2f a, b;
    a.x = arow[k + 2 * h + 0]; a.y = arow[k + 2 * h + 1];
    b.x = brow[k + 2 * h + 0]; b.y = brow[k + 2 * h + 1];
    acc = wmma4(a, b, acc);
  }
  const float scale = __expf(ls[0]);
  const float bn = Bn[n0 + r];
#pragma unroll
  for (int i = 0; i < 8; ++i) {
    int m = m0 + i + 8 * h;
    float sq = An[m] + bn - 2.0f * acc[i];
    sq = fmaxf(sq, 0.f);
    float d = (sq > 0.f) ? sqrtf(sq) : 0.f;
    K[(size_t)m * ldk + n0 + r] = __expf(-0.5f * d / scale);
  }
}

// -------------------------- per-latent Cholesky of KV = kuu + diag(V[:,l])
__global__ void chol_kernel(const float* __restrict__ kuu, const float* __restrict__ V,
                            int lbase, size_t lstride, float* __restrict__ Cbase) {
  const int l = lbase + blockIdx.x;
  float* C = Cbase + (size_t)blockIdx.x * lstride;
  const int tid = threadIdx.x, bs = blockDim.x;
  for (int idx = tid; idx < GM * GM; idx += bs) {
    int i = idx / GM, j = idx % GM;
    float v = kuu[idx];
    if (i == j) v += V[i * GL + l];
    C[idx] = v;
  }
  __syncthreads();
  for (int k = 0; k < GM; ++k) {
    if (tid == 0) C[k * GM + k] = sqrtf(C[k * GM + k]);
    __syncthreads();
    float d = C[k * GM + k];
    for (int i = k + 1 + tid; i < GM; i += bs) C[i * GM + k] /= d;
    __syncthreads();
    for (int i = k + 1 + tid; i < GM; i += bs) {
      float lik = C[i * GM + k];
      for (int j = k + 1; j <= i; ++j) C[i * GM + j] -= lik * C[j * GM + k];
    }
    __syncthreads();
  }
}

// --------------------- W = L^{-1}; thread per column, workgroup per latent (y)
__global__ void trinv_kernel(const float* __restrict__ LmBase, float* __restrict__ WBase,
                             size_t lstride) {
  const float* Lm = LmBase + (size_t)blockIdx.y * lstride;
  float* W = WBase + (size_t)blockIdx.y * lstride;
  int j = threadIdx.x;
  if (j >= GM) return;
  for (int i = 0; i < j; ++i) W[(size_t)i * GM + j] = 0.f;
  W[(size_t)j * GM + j] = 1.0f / Lm[(size_t)j * GM + j];
  for (int i = j + 1; i < GM; ++i) {
    float s = 0.f;
    for (int k = j; k < i; ++k) s += Lm[(size_t)i * GM + k] * W[(size_t)k * GM + j];
    W[(size_t)i * GM + j] = -s / Lm[(size_t)i * GM + i];
  }
}

// ---------------- Kinv = W^T * W (512^3, WMMA). Wave -> 16x64 strip (4 acc).
__global__ void ata_wmma(const float* __restrict__ WBase, float* __restrict__ KinvBase,
                         size_t lstride) {
  const float* W = WBase + (size_t)blockIdx.z * lstride;
  float* Kinv = KinvBase + (size_t)blockIdx.z * lstride;
  const int wave = threadIdx.x >> 5, lane = threadIdx.x & 31;
  const int r = lane & 15, h = lane >> 4;
  const int m0 = blockIdx.y * 16;
  const int n0 = wave * 64;
  v8f acc0 = {}, acc1 = {}, acc2 = {}, acc3 = {};
  for (int k = 0; k < GM; k += 4) {
    const float* w0 = W + (size_t)(k + 2 * h + 0) * GM;
    const float* w1 = W + (size_t)(k + 2 * h + 1) * GM;
    v2f a, b0, b1, b2, b3;
    a.x  = w0[m0 + r];      a.y  = w1[m0 + r];
    b0.x = w0[n0 + r];      b0.y = w1[n0 + r];
    b1.x = w0[n0 + 16 + r]; b1.y = w1[n0 + 16 + r];
    b2.x = w0[n0 + 32 + r]; b2.y = w1[n0 + 32 + r];
    b3.x = w0[n0 + 48 + r]; b3.y = w1[n0 + 48 + r];
    acc0 = wmma4(a, b0, acc0);
    acc1 = wmma4(a, b1, acc1);
    acc2 = wmma4(a, b2, acc2);
    acc3 = wmma4(a, b3, acc3);
  }
#pragma unroll
  for (int i = 0; i < 8; ++i) {
    size_t row = (size_t)(m0 + i + 8 * h) * GM;
    Kinv[row + n0 +  0 + r] = acc0[i];
    Kinv[row + n0 + 16 + r] = acc1[i];
    Kinv[row + n0 + 32 + r] = acc2[i];
    Kinv[row + n0 + 48 + r] = acc3[i];
  }
}

// ----- T = kxu * Kinv (4096x512x512), LDS-tiled with async global->LDS DMA.
// Block: 64m x 128n, 8 waves (4x2), wave = 16m x 64n (4 WMMA accumulators).
// Kinv is exactly symmetric (W^T W elementwise-identical), so B[k][n] is read
// as Kinv[n][k]: both A and B chunks are contiguous row segments.
__global__ void __launch_bounds__(256)
ab_wmma_lds(const float* __restrict__ A, const float* __restrict__ Bsym,
            float* __restrict__ C) {
  __shared__ float tileA[64 * LDSTR];    // 64 m-rows x KC
  __shared__ float tileB[128 * LDSTR];   // 128 n-rows x KC

  const int tid = threadIdx.x;
  const int wave = tid >> 5, lane = tid & 31;
  const int r = lane & 15, h = lane >> 4;
  const int wm = wave >> 1;              // 0..3
  const int wn = wave & 1;               // 0..1
  const int mblk = blockIdx.y * 64;
  const int nblk = blockIdx.x * 128;

  // Low 32 bits of a flat LDS pointer are the LDS byte offset (ISA 10.2).
  const uint32_t ldsA = (uint32_t)(uintptr_t)tileA;
  const uint32_t ldsB = (uint32_t)(uintptr_t)tileB;

  const int srow = tid >> 4;             // 0..15 staging row
  const int scol = (tid & 15) * 4;       // dword column 0..60

  const float* arow = tileA + (wm * 16 + r) * LDSTR;
  const float* br0 = tileB + (wn * 64 +  0 + r) * LDSTR;
  const float* br1 = tileB + (wn * 64 + 16 + r) * LDSTR;
  const float* br2 = tileB + (wn * 64 + 32 + r) * LDSTR;
  const float* br3 = tileB + (wn * 64 + 48 + r) * LDSTR;

  v8f acc0 = {}, acc1 = {}, acc2 = {}, acc3 = {};

  for (int kk = 0; kk < GM; kk += KC) {
    // ---- async DMA the A (64 rows) and B (128 rows) k-chunk into LDS ----
#pragma unroll
    for (int j = 0; j < 4; ++j) {
      int row = srow + j * 16;
      async_copy_b128(ldsA + (uint32_t)(row * LDSTR + scol) * 4u,
                      A + (size_t)(mblk + row) * GM + kk + scol);
    }
#pragma unroll
    for (int j = 0; j < 8; ++j) {
      int row = srow + j * 16;
      async_copy_b128(ldsB + (uint32_t)(row * LDSTR + scol) * 4u,
                      Bsym + (size_t)(nblk + row) * GM + kk + scol);
    }
    async_wait0();
    __syncthreads();

    // ---- consume the chunk: 16 k-steps, 4 WMMAs each ----
#pragma unroll
    for (int k = 0; k < KC; k += 4) {
      int k0 = k + 2 * h;
      v2f a, b0, b1, b2, b3;
      a.x  = arow[k0]; a.y  = arow[k0 + 1];
      b0.x = br0[k0];  b0.y = br0[k0 + 1];
      b1.x = br1[k0];  b1.y = br1[k0 + 1];
      b2.x = br2[k0];  b2.y = br2[k0 + 1];
      b3.x = br3[k0];  b3.y = br3[k0 + 1];
      acc0 = wmma4(a, b0, acc0);
      acc1 = wmma4(a, b1, acc1);
      acc2 = wmma4(a, b2, acc2);
      acc3 = wmma4(a, b3, acc3);
    }
    __syncthreads();
  }

#pragma unroll
  for (int i = 0; i < 8; ++i) {
    size_t row = (size_t)(mblk + wm * 16 + i + 8 * h) * GM;
    int nb = nblk + wn * 64 + r;
    C[row + nb +  0] = acc0[i];
    C[row + nb + 16] = acc1[i];
    C[row + nb + 32] = acc2[i];
    C[row + nb + 48] = acc3[i];
  }
}

// ----------------------- alpha_l = Kinv_l * Y[:, l]; blockIdx.y = latent slot
__global__ void gemv_kernel(const float* __restrict__ KinvBase, const float* __restrict__ Y,
                            int lbase, size_t lstride, float* __restrict__ alphaBase,
                            int astride) {
  const int slot = blockIdx.y;
  const int l = lbase + slot;
  const float* Kinv = KinvBase + (size_t)slot * lstride;
  float* alpha = alphaBase + (size_t)slot * astride;
  int m = blockIdx.x * blockDim.x + threadIdx.x;
  if (m >= GM) return;
  float s = 0.f;
  for (int k = 0; k < GM; ++k) s += Kinv[(size_t)m * GM + k] * Y[k * GL + l];
  alpha[m] = s;
}

// ------------- mu[n,l] = kxu[n,:].alpha ; var[n,l] = 1 - kxu[n,:].T[n,:]
__global__ void finalize_kernel(const float* __restrict__ kxu, const float* __restrict__ T,
                                const float* __restrict__ alpha, int l,
                                float* __restrict__ out) {
  int n = blockIdx.x * blockDim.x + threadIdx.x;
  if (n >= GN) return;
  const float* kr = kxu + (size_t)n * GM;
  const float* tr = T + (size_t)n * GM;
  float smu = 0.f, sq = 0.f;
  for (int m = 0; m < GM; ++m) {
    float kx = kr[m];
    smu += kx * alpha[m];
    sq  += kx * tr[m];
  }
  out[(size_t)n * GL + l] = smu;                              // mu
  out[(size_t)GN * GL + (size_t)n * GL + l] = 1.0f - sq;      // var
}

extern "C" void kernel_launch(void* const* d_in, const int* in_sizes, int n_in,
                              void* d_out, int out_size, void* d_ws, size_t ws_size,
                              hipStream_t stream) {
  const float* x  = (const float*)d_in[0];   // (N, D)
  const float* U  = (const float*)d_in[1];   // (M, D)
  const float* Y  = (const float*)d_in[2];   // (M, L)
  const float* V  = (const float*)d_in[3];   // (M, L)
  const float* ls = (const float*)d_in[4];   // (1,)
  float* out = (float*)d_out;                // mu (N,L) then var (N,L)

  const size_t MM = (size_t)GM * GM;
  const size_t NM = (size_t)GN * GM;

  const size_t batched_floats = NM + MM + 3 * (size_t)GL * MM + NM
                              + (size_t)GL * GM + GN + GM;
  const bool batched = ws_size >= batched_floats * sizeof(float);

  float* kxu = (float*)d_ws;                 // N*M
  float* kuu = kxu + NM;                     // M*M
  float* chol = kuu + MM;                    // M*M  (x GL if batched)
  float* linv = chol + (batched ? (size_t)GL * MM : MM);
  float* kinv = linv + (batched ? (size_t)GL * MM : MM);
  float* T    = kinv + (batched ? (size_t)GL * MM : MM);   // N*M (shared)
  float* alpha = T + NM;                     // GM (x GL if batched)
  float* xn = alpha + (batched ? (size_t)GL * GM : (size_t)GM);
  float* un = xn + GN;

  row_norms<<<dim3((GN + 255) / 256), 256, 0, stream>>>(x, xn, GN);
  row_norms<<<dim3((GM + 255) / 256), 256, 0, stream>>>(U, un, GM);
  rbf_wmma<<<dim3(GM / 128, GN / 16), 256, 0, stream>>>(x, xn, U, un, ls, kxu, GM);
  rbf_wmma<<<dim3(GM / 128, GM / 16), 256, 0, stream>>>(U, un, U, un, ls, kuu, GM);

  if (batched) {
    chol_kernel<<<dim3(GL), 512, 0, stream>>>(kuu, V, 0, MM, chol);
    trinv_kernel<<<dim3(1, GL), 512, 0, stream>>>(chol, linv, MM);
    ata_wmma<<<dim3(1, GM / 16, GL), 256, 0, stream>>>(linv, kinv, MM);
    gemv_kernel<<<dim3(2, GL), 256, 0, stream>>>(kinv, Y, 0, MM, alpha, GM);
    for (int l = 0; l < GL; ++l) {
      ab_wmma_lds<<<dim3(GM / 128, GN / 64), 256, 0, stream>>>(
          kxu, kinv + (size_t)l * MM, T);
      finalize_kernel<<<dim3((GN + 255) / 256), 256, 0, stream>>>(
          kxu, T, alpha + (size_t)l * GM, l, out);
    }
  } else {
    for (int l = 0; l < GL; ++l) {
      chol_kernel<<<dim3(1), 512, 0, stream>>>(kuu, V, l, 0, chol);
      trinv_kernel<<<dim3(1, 1), 512, 0, stream>>>(chol, linv, 0);
      ata_wmma<<<dim3(1, GM / 16, 1), 256, 0, stream>>>(linv, kinv, 0);
      gemv_kernel<<<dim3(2, 1), 256, 0, stream>>>(kinv, Y, l, 0, alpha, 0);
      ab_wmma_lds<<<dim3(GM / 128, GN / 64), 256, 0, stream>>>(kxu, kinv, T);
      finalize_kernel<<<dim3((GN + 255) / 256), 256, 0, stream>>>(kxu, T, alpha, l, out);
    }
  }
}